// SG_21543555957000
// MI455X (gfx1250) — compile-verified
//
#include <hip/hip_runtime.h>

typedef __attribute__((ext_vector_type(2))) float v2f;
typedef __attribute__((ext_vector_type(8))) float v8f;

// ---------------------------------------------------------------- degrees ---
__global__ void k_deg_init(float* __restrict__ deg, int n) {
    int i = blockIdx.x * blockDim.x + threadIdx.x;
    if (i < n) deg[i] = 1.0f;               // self-loop contributes 1
}

__global__ void k_deg_accum(const int* __restrict__ dst, float* __restrict__ deg, int E) {
    int i = blockIdx.x * blockDim.x + threadIdx.x;
    int stride = gridDim.x * blockDim.x;
    for (; i < E; i += stride) atomicAdd(&deg[dst[i]], 1.0f);
}

__global__ void k_rsqrt_inplace(float* __restrict__ d, int n) {
    int i = blockIdx.x * blockDim.x + threadIdx.x;
    if (i < n) {
        float v = d[i];
        d[i] = (v > 0.0f) ? rsqrtf(v) : 0.0f;
    }
}

// ----------------------------------------------------- aggregation (edges) ---
// agg[i,:] = x[i,:] * dinv[i]^2   (self-loop term; doubles as the zero-init)
template <int F>
__global__ void k_agg_init(const float* __restrict__ x, const float* __restrict__ dinv,
                           float* __restrict__ agg, int n) {
    int tid = blockIdx.x * blockDim.x + threadIdx.x;
    int total = n * (F / 4);
    if (tid >= total) return;
    int node = tid / (F / 4);
    float di = dinv[node];
    float s = di * di;
    const float4 v = *(const float4*)(x + (size_t)tid * 4);
    float4 r = make_float4(v.x * s, v.y * s, v.z * s, v.w * s);
    *(float4*)(agg + (size_t)tid * 4) = r;
}

// agg[dst,:] += x[src,:] * dinv[src]*dinv[dst]   over all directed edges.
// F/4 threads per edge; float4 gather + 4 f32 atomics (resolve in L2 —
// a 100K x 64 f32 feature matrix is 25.6MB, well inside the 192MB L2).
template <int F>
__global__ void k_agg_edges(const float* __restrict__ x, const int* __restrict__ src,
                            const int* __restrict__ dst, const float* __restrict__ dinv,
                            float* __restrict__ agg, int E) {
    constexpr int TPE = F / 4;
    int tid = blockIdx.x * blockDim.x + threadIdx.x;
    int e  = tid / TPE;
    int f4 = (tid % TPE) * 4;
    int estride = (gridDim.x * blockDim.x) / TPE;
    for (; e < E; e += estride) {
        int s = src[e];
        int d = dst[e];
        float nrm = dinv[s] * dinv[d];
        const float4 v = *(const float4*)(x + (size_t)s * F + f4);
        float* o = agg + (size_t)d * F + f4;
        atomicAdd(o + 0, v.x * nrm);
        atomicAdd(o + 1, v.y * nrm);
        atomicAdd(o + 2, v.z * nrm);
        atomicAdd(o + 3, v.w * nrm);
    }
}

// --------------------------------------------------------- WMMA fp32 GEMM ---
// out[n, o] = relu?( sum_k agg[n,k] * W[o,k] + b[o] )
// One wave per 16-row tile. V_WMMA_F32_16X16X4_F32, K swept 4 at a time.
// A 16x4 layout: lane l -> row (l&15), K pair (l>>4)*2.  B mirrors with M<->N.
// C/D 16x16 f32: VGPR r, lanes 0-15 -> M=r, lanes 16-31 -> M=r+8.
template <int Fin, int Fout, bool RELU>
__global__ void k_gemm_wmma(const float* __restrict__ agg, const float* __restrict__ W,
                            const float* __restrict__ bias, float* __restrict__ out,
                            int nrows) {
    constexpr int FoutPad = (Fout + 15) & ~15;
    int wave = (blockIdx.x * blockDim.x + threadIdx.x) >> 5;
    int lane = threadIdx.x & 31;
    int row0 = wave * 16;
    if (row0 >= nrows) return;                 // wave-uniform: EXEC stays all-1s

    int lm = lane & 15;
    int hi = lane >> 4;                        // 0 or 1
    int arow = row0 + lm;
    if (arow > nrows - 1) arow = nrows - 1;    // clamp (keeps EXEC full)
    const bool fullrows = (row0 + 16 <= nrows);

#pragma unroll
    for (int nt = 0; nt < FoutPad / 16; ++nt) {
        int o = nt * 16 + lm;
        bool ovalid = (FoutPad == Fout) || (o < Fout);
        float bv = ovalid ? bias[o] : 0.0f;
        v8f c;
#pragma unroll
        for (int r = 0; r < 8; ++r) c[r] = bv;

#pragma unroll
        for (int kt = 0; kt < Fin / 4; ++kt) {
            int k0 = kt * 4 + hi * 2;
            v2f a, bm;
            a.x = agg[(size_t)arow * Fin + k0];
            a.y = agg[(size_t)arow * Fin + k0 + 1];
            bm.x = ovalid ? W[(size_t)o * Fin + k0]     : 0.0f;
            bm.y = ovalid ? W[(size_t)o * Fin + k0 + 1] : 0.0f;
            c = __builtin_amdgcn_wmma_f32_16x16x4_f32(
                    false, a, false, bm, (short)0, c, false, false);
        }

        if constexpr (FoutPad == Fout) {
            if (fullrows) {
                // branch-free fast path: 8 stores at immediate offsets
                float* op = out + (size_t)(row0 + hi * 8) * Fout + o;
#pragma unroll
                for (int r = 0; r < 8; ++r) {
                    float v = c[r];
                    if (RELU) v = fmaxf(v, 0.0f);
                    op[(size_t)r * Fout] = v;
                }
            } else {
#pragma unroll
                for (int r = 0; r < 8; ++r) {
                    int node = row0 + r + hi * 8;
                    float v = c[r];
                    if (RELU) v = fmaxf(v, 0.0f);
                    if (node < nrows) out[(size_t)node * Fout + o] = v;
                }
            }
        } else {
#pragma unroll
            for (int r = 0; r < 8; ++r) {
                int node = row0 + r + hi * 8;
                float v = c[r];
                if (RELU) v = fmaxf(v, 0.0f);
                if (ovalid && node < nrows) out[(size_t)node * Fout + o] = v;
            }
        }
    }
}

// -------------------------------------------------------- final 10->2 head ---
__global__ void k_final(const float* __restrict__ h, const float* __restrict__ Wl,
                        const float* __restrict__ bl, float* __restrict__ out, int n) {
    int i = blockIdx.x * blockDim.x + threadIdx.x;
    if (i >= n) return;
    float a0 = bl[0], a1 = bl[1];
#pragma unroll
    for (int k = 0; k < 10; ++k) {
        float v = h[(size_t)i * 10 + k];
        a0 += v * Wl[k];
        a1 += v * Wl[10 + k];
    }
    out[(size_t)i * 2 + 0] = a0;
    out[(size_t)i * 2 + 1] = a1;
}

// -------------------------------------------------------------------- host ---
static inline int edge_blocks(long long work) {
    long long b = (work + 255) / 256;
    return (int)(b > 16384 ? 16384 : b);
}

extern "C" void kernel_launch(void* const* d_in, const int* in_sizes, int n_in,
                              void* d_out, int out_size, void* d_ws, size_t ws_size,
                              hipStream_t stream) {
    const float* x  = (const float*)d_in[0];
    const int*   ei = (const int*)d_in[1];
    const float* W1 = (const float*)d_in[2];  const float* b1 = (const float*)d_in[3];
    const float* W2 = (const float*)d_in[4];  const float* b2 = (const float*)d_in[5];
    const float* W3 = (const float*)d_in[6];  const float* b3 = (const float*)d_in[7];
    const float* W4 = (const float*)d_in[8];  const float* b4 = (const float*)d_in[9];
    const float* Wl = (const float*)d_in[10]; const float* bl = (const float*)d_in[11];

    const int N = in_sizes[0] / 8;
    const int E = in_sizes[1] / 2;
    const int* src = ei;
    const int* dst = ei + E;

    float* ws   = (float*)d_ws;
    float* dinv = ws;                       // N floats (deg, then rsqrt in place)
    float* bufA = ws + N;                   // N*64 floats (activations)
    float* bufB = bufA + (size_t)N * 64;    // N*64 floats (aggregation scratch)

    const int nb = (N + 255) / 256;
    const int gemm_blocks = (((N + 15) / 16) + 7) / 8;   // 8 waves per 256-thr block

    // degrees -> dinv
    k_deg_init<<<nb, 256, 0, stream>>>(dinv, N);
    k_deg_accum<<<edge_blocks(E), 256, 0, stream>>>(dst, dinv, E);
    k_rsqrt_inplace<<<nb, 256, 0, stream>>>(dinv, N);

    // layer 1: 8 -> 16, relu
    k_agg_init<8><<<(N * 2 + 255) / 256, 256, 0, stream>>>(x, dinv, bufB, N);
    k_agg_edges<8><<<edge_blocks((long long)E * 2), 256, 0, stream>>>(x, src, dst, dinv, bufB, E);
    k_gemm_wmma<8, 16, true><<<gemm_blocks, 256, 0, stream>>>(bufB, W1, b1, bufA, N);

    // layer 2: 16 -> 64, relu
    k_agg_init<16><<<(N * 4 + 255) / 256, 256, 0, stream>>>(bufA, dinv, bufB, N);
    k_agg_edges<16><<<edge_blocks((long long)E * 4), 256, 0, stream>>>(bufA, src, dst, dinv, bufB, E);
    k_gemm_wmma<16, 64, true><<<gemm_blocks, 256, 0, stream>>>(bufB, W2, b2, bufA, N);

    // layer 3: 64 -> 64, relu
    k_agg_init<64><<<(N * 16 + 255) / 256, 256, 0, stream>>>(bufA, dinv, bufB, N);
    k_agg_edges<64><<<edge_blocks((long long)E * 16), 256, 0, stream>>>(bufA, src, dst, dinv, bufB, E);
    k_gemm_wmma<64, 64, true><<<gemm_blocks, 256, 0, stream>>>(bufB, W3, b3, bufA, N);

    // layer 4: 64 -> 10, no relu
    k_agg_init<64><<<(N * 16 + 255) / 256, 256, 0, stream>>>(bufA, dinv, bufB, N);
    k_agg_edges<64><<<edge_blocks((long long)E * 16), 256, 0, stream>>>(bufA, src, dst, dinv, bufB, E);
    k_gemm_wmma<64, 10, false><<<gemm_blocks, 256, 0, stream>>>(bufB, W4, b4, bufA, N);

    // head: 10 -> 2
    k_final<<<nb, 256, 0, stream>>>(bufA, Wl, bl, (float*)d_out, N);
}